// DyRepClassifier_79663053406463
// MI455X (gfx1250) — compile-verified
//
#include <hip/hip_runtime.h>
#include <hip/hip_bf16.h>

typedef float v2f __attribute__((ext_vector_type(2)));
typedef float v8f __attribute__((ext_vector_type(8)));

#define HD   128
#define RAWD 100
#define FD   172
#define CD   50
#define GATES 384

// ---- workspace layout (bytes) ----
// [0 .. 384*4)      : gi[384]
// [384*4]           : dec_touched
// [385*4]           : relu(decay)
// [386*4]           : current_time (float)
// [2048 .. 2048+4N) : mark[N] (int)

__global__ void dyrep_event_precompute(const float* __restrict__ t,
                                       const float* __restrict__ msg,
                                       const float* __restrict__ W_msg,
                                       const float* __restrict__ b_msg,
                                       const float* __restrict__ W_ih,
                                       const float* __restrict__ b_ih,
                                       const float* __restrict__ decay,
                                       const int*   __restrict__ current_time,
                                       float* __restrict__ ws) {
  __shared__ float em[HD];
  int tid = threadIdx.x;  // 128 threads
  float acc = b_msg[tid];
  const float* wrow = W_msg + tid * RAWD;
  for (int k = 0; k < RAWD; ++k) acc += wrow[k] * msg[k];
  em[tid] = acc;
  __syncthreads();
  for (int j = tid; j < GATES; j += 128) {
    float g = b_ih[j];
    const float* w = W_ih + j * HD;
    for (int k = 0; k < HD; ++k) g += w[k] * em[k];
    ws[j] = g;
  }
  if (tid == 0) {
    float dr = decay[0] > 0.f ? decay[0] : 0.f;
    float ct = (float)current_time[0];
    float et = t[0];
    ws[384] = expf(-dr * (ct - et));
    ws[385] = dr;
    ws[386] = ct;
  }
}

__global__ void dyrep_zero_mark(int* __restrict__ mark, int n) {
  int i = blockIdx.x * blockDim.x + threadIdx.x;
  if (i < n) mark[i] = 0;
}

__global__ void dyrep_mark(const int* __restrict__ src, const int* __restrict__ dst,
                           int e, int* __restrict__ mark) {
  int i = blockIdx.x * blockDim.x + threadIdx.x;
  if (i < e) { mark[src[i]] = 1; mark[dst[i]] = 1; }
}

// LDS row strides (padded for bank spread; even so float2 loads stay 8B-aligned)
#define SS  132   // node_state tile stride
#define NS  178   // node_features tile stride
#define BS  132   // blended tile stride
#define HS  132   // h1 tile stride

__global__ __launch_bounds__(256)
void dyrep_main(const int*   __restrict__ label_nodes,
                const float* __restrict__ memory_buf,
                const float* __restrict__ node_state,
                const float* __restrict__ last_seen,
                const float* __restrict__ node_features,
                const float* __restrict__ W_hh,  const float* __restrict__ b_hh,
                const float* __restrict__ W_feat,const float* __restrict__ b_feat,
                const float* __restrict__ W1,    const float* __restrict__ b1,
                const float* __restrict__ W2,    const float* __restrict__ b2,
                const float* __restrict__ ws,    const int* __restrict__ mark,
                float* __restrict__ out, int Bn) {
  __shared__ float Slds [16 * SS];
  __shared__ float NFlds[16 * NS];
  __shared__ float BLlds[16 * BS];
  __shared__ float H1lds[16 * HS];
  __shared__ float gilds[GATES];
  __shared__ int   u16[16];
  __shared__ int   mk16[16];
  __shared__ float dec16[16];

  const int tid  = threadIdx.x;
  const int wave = tid >> 5;
  const int lane = tid & 31;
  const int ln   = lane & 15;   // A-row (label) / B-col index
  const int ch   = lane >> 4;   // K-pair selector (0: k0,k0+1 ; 1: k0+2,k0+3)
  const int base = blockIdx.x * 16;

  // ---- phase A: per-label metadata + shared gi ----
  if (tid < 16) {
    int li = base + tid; if (li >= Bn) li = Bn - 1;
    int u = label_nodes[li];
    u16[tid] = u;
    int mkd = mark[u];
    mk16[tid] = mkd;
    float dt = ws[386] - last_seen[u];
    dec16[tid] = mkd ? ws[384] : expf(-ws[385] * dt);
  }
  for (int j = tid; j < GATES; j += 256) gilds[j] = ws[j];
  __syncthreads();

  // ---- phase B: stage node_state + node_features tiles into LDS ----
  {
    int row = tid >> 4, c0 = (tid & 15) * 8;
    const float* sp = node_state + (size_t)u16[row] * HD + c0;
    float* dp = Slds + row * SS + c0;
#pragma unroll
    for (int k = 0; k < 8; ++k) dp[k] = sp[k];
  }
  {
    int row = tid >> 4;
    const float* sp = node_features + (size_t)u16[row] * FD;
    float* dp = NFlds + row * NS;
    for (int c = tid & 15; c < FD; c += 16) dp[c] = sp[c];
  }
  __syncthreads();

  // ---- phase C: GRU gate GEMMs (W_hh) + feature GEMM (W_feat) + blend ----
  const int hcol = wave * 16 + ln;   // h index this wave/lane owns as output col
  v8f dr, dz, dn, df;
  {
    float br = b_hh[hcol], bz = b_hh[128 + hcol], bnn = b_hh[256 + hcol];
    float bf = b_feat[hcol];
#pragma unroll
    for (int v = 0; v < 8; ++v) { dr[v] = br; dz[v] = bz; dn[v] = bnn; df[v] = bf; }
  }
#pragma unroll 4
  for (int k0 = 0; k0 < HD; k0 += 4) {
    v2f a  = *(const v2f*)(Slds + ln * SS + k0 + 2 * ch);
    v2f br = *(const v2f*)(W_hh + (size_t)(0   + hcol) * HD + k0 + 2 * ch);
    v2f bz = *(const v2f*)(W_hh + (size_t)(128 + hcol) * HD + k0 + 2 * ch);
    v2f bn = *(const v2f*)(W_hh + (size_t)(256 + hcol) * HD + k0 + 2 * ch);
    dr = __builtin_amdgcn_wmma_f32_16x16x4_f32(false, a, false, br, (short)0, dr, false, false);
    dz = __builtin_amdgcn_wmma_f32_16x16x4_f32(false, a, false, bz, (short)0, dz, false, false);
    dn = __builtin_amdgcn_wmma_f32_16x16x4_f32(false, a, false, bn, (short)0, dn, false, false);
  }
#pragma unroll 4
  for (int k0 = 0; k0 < FD; k0 += 4) {   // FD = 172 = 43 * 4
    v2f a = *(const v2f*)(NFlds + ln * NS + k0 + 2 * ch);
    v2f b = *(const v2f*)(W_feat + (size_t)hcol * FD + k0 + 2 * ch);
    df = __builtin_amdgcn_wmma_f32_16x16x4_f32(false, a, false, b, (short)0, df, false, false);
  }
  {
    float gir = gilds[hcol], giz = gilds[128 + hcol], gin = gilds[256 + hcol];
#pragma unroll
    for (int v = 0; v < 8; ++v) {
      int m = v + 8 * ch;                   // label row within tile
      float prev = Slds[m * SS + hcol];
      float r  = 1.f / (1.f + expf(-(gir + dr[v])));
      float z  = 1.f / (1.f + expf(-(giz + dz[v])));
      float nn = tanhf(gin + r * dn[v]);
      float sp = mk16[m] ? ((1.f - z) * nn + z * prev) : prev;
      float bl = sp * dec16[m] + df[v] + memory_buf[(size_t)u16[m] * HD + hcol];
      BLlds[m * BS + hcol] = bl;
    }
  }
  __syncthreads();

  // ---- phase D: h1 = relu(BL @ W1^T + b1) ----
  v8f d1;
  { float bb = b1[hcol];
#pragma unroll
    for (int v = 0; v < 8; ++v) d1[v] = bb; }
#pragma unroll 4
  for (int k0 = 0; k0 < HD; k0 += 4) {
    v2f a = *(const v2f*)(BLlds + ln * BS + k0 + 2 * ch);
    v2f b = *(const v2f*)(W1 + (size_t)hcol * HD + k0 + 2 * ch);
    d1 = __builtin_amdgcn_wmma_f32_16x16x4_f32(false, a, false, b, (short)0, d1, false, false);
  }
#pragma unroll
  for (int v = 0; v < 8; ++v) {
    int m = v + 8 * ch;
    float h1 = d1[v] > 0.f ? d1[v] : 0.f;
    H1lds[m * HS + hcol] = h1;
  }
  __syncthreads();

  // ---- phase E: out = H1 @ W2^T + b2 (waves 0..3 cover 50 cols) ----
  if (wave < 4) {
    int col = wave * 16 + ln;
    bool valid = col < CD;
    int wrow = valid ? col : (CD - 1);
    v8f d2;
    { float bb = valid ? b2[col] : 0.f;
#pragma unroll
      for (int v = 0; v < 8; ++v) d2[v] = bb; }
#pragma unroll 4
    for (int k0 = 0; k0 < HD; k0 += 4) {
      v2f a = *(const v2f*)(H1lds + ln * HS + k0 + 2 * ch);
      v2f b = *(const v2f*)(W2 + (size_t)wrow * HD + k0 + 2 * ch);
      d2 = __builtin_amdgcn_wmma_f32_16x16x4_f32(false, a, false, b, (short)0, d2, false, false);
    }
    if (valid) {
#pragma unroll
      for (int v = 0; v < 8; ++v) {
        int m = v + 8 * ch;
        int li = base + m;
        if (li < Bn) out[(size_t)li * CD + col] = d2[v];
      }
    }
  }
}

extern "C" void kernel_launch(void* const* d_in, const int* in_sizes, int n_in,
                              void* d_out, int out_size, void* d_ws, size_t ws_size,
                              hipStream_t stream) {
  const int*   label_nodes   = (const int*)  d_in[0];
  const int*   src           = (const int*)  d_in[1];
  const int*   dst           = (const int*)  d_in[2];
  const float* t             = (const float*)d_in[3];
  const float* msg           = (const float*)d_in[4];
  const float* memory_buf    = (const float*)d_in[5];
  const float* node_state    = (const float*)d_in[6];
  const float* last_seen     = (const float*)d_in[7];
  const float* node_features = (const float*)d_in[8];
  const float* decay         = (const float*)d_in[9];
  const float* W_msg         = (const float*)d_in[10];
  const float* b_msg         = (const float*)d_in[11];
  const float* W_ih          = (const float*)d_in[12];
  const float* W_hh          = (const float*)d_in[13];
  const float* b_ih          = (const float*)d_in[14];
  const float* b_hh          = (const float*)d_in[15];
  const float* W_feat        = (const float*)d_in[16];
  const float* b_feat        = (const float*)d_in[17];
  const float* W1            = (const float*)d_in[18];
  const float* b1            = (const float*)d_in[19];
  const float* W2            = (const float*)d_in[20];
  const float* b2            = (const float*)d_in[21];
  const int*   current_time  = (const int*)  d_in[22];

  int Bn = in_sizes[0];     // 200000 labels
  int E  = in_sizes[1];     // 20000 events
  int N  = in_sizes[7];     // 500000 nodes (last_seen length)

  float* wsf  = (float*)d_ws;
  int*   mark = (int*)((char*)d_ws + 2048);

  dyrep_event_precompute<<<1, 128, 0, stream>>>(t, msg, W_msg, b_msg, W_ih, b_ih,
                                                decay, current_time, wsf);
  dyrep_zero_mark<<<(N + 255) / 256, 256, 0, stream>>>(mark, N);
  dyrep_mark<<<(E + 255) / 256, 256, 0, stream>>>(src, dst, E, mark);

  int tiles = (Bn + 15) / 16;
  dyrep_main<<<tiles, 256, 0, stream>>>(label_nodes, memory_buf, node_state, last_seen,
                                        node_features, W_hh, b_hh, W_feat, b_feat,
                                        W1, b1, W2, b2, wsf, mark, (float*)d_out, Bn);
}